// GLCN_4879082848784
// MI455X (gfx1250) — compile-verified
//
#include <hip/hip_runtime.h>
#include <hip/hip_bf16.h>
#include <math.h>

typedef __attribute__((ext_vector_type(2))) float v2f;
typedef __attribute__((ext_vector_type(8))) float v8f;

#define BATCH 32
#define NNODE 512
#define FDIM  320
#define FOBS  256
#define DK    64
#define MROWS (BATCH * NNODE)          // 16384 flattened rows
#define HI_ELEMS ((size_t)MROWS * DK)  // floats per projection buffer

// ---------------------------------------------------------------------------
// Kernel 1: projections  P = hx @ W   (W = a_link or b_link, 256x64)
// One wave per (16-row tile, which-W). 2048 waves = 256 blocks x 8 waves.
// Each wave computes a 16x64 strip: 4 f32 accumulators, 64 chained
// v_wmma_f32_16x16x4_f32 steps, A-fragment reused across the 4 B tiles.
// ---------------------------------------------------------------------------
__global__ __launch_bounds__(256) void glcn_proj_kernel(
    const float* __restrict__ h,
    const float* __restrict__ a_link,
    const float* __restrict__ b_link,
    float* __restrict__ ws) {
  const int lane = threadIdx.x & 31;
  const int w = blockIdx.x * 8 + (threadIdx.x >> 5);  // 0..2047
  const int which = w & 1;
  const int rowTile = w >> 1;                         // 0..1023
  const int i_base = rowTile * 16;                    // flattened (b*N + n) row
  const float* W = which ? b_link : a_link;
  float* out = ws + (which ? HI_ELEMS : 0);

  // 16x4 f32 A layout: lane = m + 16*(k/2), reg = k&1  (K pair contiguous -> b64)
  const int m = lane & 15;
  const int koff = (lane >> 4) * 2;

  v8f acc[4] = {};
  const float* arow = h + (size_t)(i_base + m) * FDIM;  // only first 256 cols used
  for (int kk = 0; kk < FOBS / 4; ++kk) {
    const int k0 = kk * 4;
    v2f a = *(const v2f*)(arow + k0 + koff);
#pragma unroll
    for (int nt = 0; nt < 4; ++nt) {
      // 4x16 B layout mirrors A: lane = n + 16*(k/2), reg = k&1.
      // W is (256 x 64) row-major: element (k, n) at W[k*64 + n] (stride 64 apart).
      const float* bp = W + (size_t)(k0 + koff) * DK + nt * 16 + m;
      v2f b;
      b.x = bp[0];
      b.y = bp[DK];
      acc[nt] = __builtin_amdgcn_wmma_f32_16x16x4_f32(
          false, a, false, b, (short)0, acc[nt], false, false);
    }
  }

  // C/D 16x16 f32 layout: reg r -> M = r + 8*(lane>=16); N = lane&15
#pragma unroll
  for (int nt = 0; nt < 4; ++nt) {
#pragma unroll
    for (int r = 0; r < 8; ++r) {
      const int mi = r + 8 * (lane >> 4);
      const int nj = nt * 16 + (lane & 15);
      out[(size_t)(i_base + mi) * DK + nj] = acc[nt][r];
    }
  }
}

// ---------------------------------------------------------------------------
// Kernel 2: logits = h_i @ h_j^T / 8, fused sigmoid/threshold/log epilogue.
// One wave per 16x64 tile of each 512x512 batch matrix:
// 32 i-tiles x 8 j-strips x 32 batches = 8192 waves = 1024 blocks x 8 waves.
// B = h_j^T so the B fragment's (k, k+1) pair is contiguous in h_j -> b64 load.
// ---------------------------------------------------------------------------
__global__ __launch_bounds__(256) void glcn_logits_kernel(
    const float* __restrict__ ws,
    float* __restrict__ Aout,
    float* __restrict__ partials) {
  const int lane = threadIdx.x & 31;
  const int w = blockIdx.x * 8 + (threadIdx.x >> 5);  // 0..8191
  const int b = w >> 8;
  const int rem = w & 255;
  const int i_base = (rem >> 3) * 16;
  const int j_base = (rem & 7) * 64;

  const float* hi = ws + (size_t)b * NNODE * DK;
  const float* hj = ws + HI_ELEMS + (size_t)b * NNODE * DK;

  const int m = lane & 15;
  const int koff = (lane >> 4) * 2;

  v8f acc[4] = {};
  const float* arow = hi + (size_t)(i_base + m) * DK;
  for (int kk = 0; kk < DK / 4; ++kk) {
    const int k0 = kk * 4;
    v2f a = *(const v2f*)(arow + k0 + koff);
#pragma unroll
    for (int nt = 0; nt < 4; ++nt) {
      // B[k][n] = h_j[b, j_base+nt*16+n, k]; (k, k+1) contiguous along last dim
      v2f bb = *(const v2f*)(hj + (size_t)(j_base + nt * 16 + m) * DK + k0 + koff);
      acc[nt] = __builtin_amdgcn_wmma_f32_16x16x4_f32(
          false, a, false, bb, (short)0, acc[nt], false, false);
    }
  }

  float lsum = 0.0f;
  float* outb = Aout + (size_t)b * NNODE * NNODE;
#pragma unroll
  for (int nt = 0; nt < 4; ++nt) {
#pragma unroll
    for (int r = 0; r < 8; ++r) {
      const float logit = acc[nt][r] * 0.125f;  // 1/sqrt(D_K)
      const int i = i_base + r + 8 * (lane >> 4);
      const int j = j_base + nt * 16 + (lane & 15);
      const float ys = 1.0f / (1.0f + __expf(-logit));
      const bool hard = logit > 0.0f;            // sigmoid(x) > 0.5  <=>  x > 0
      const float sel = hard ? ys : (1.0f - ys);
      const float lp = __logf(sel + 1e-8f);
      const bool diag = (i == j);
      outb[(size_t)i * NNODE + j] = diag ? 1.0f : (hard ? 1.0f : 0.0f);
      lsum += diag ? 0.0f : lp;
    }
  }
  // deterministic wave32 reduction
#pragma unroll
  for (int off = 16; off >= 1; off >>= 1)
    lsum += __shfl_xor(lsum, off, 32);
  if (lane == 0) partials[w] = lsum;
}

// ---------------------------------------------------------------------------
// Kernel 3: deterministic fixed-order tree reduction of 256 partials per batch.
// ---------------------------------------------------------------------------
__global__ __launch_bounds__(256) void glcn_probs_kernel(
    const float* __restrict__ partials, float* __restrict__ probs) {
  __shared__ float s[256];
  const int b = blockIdx.x;
  const int t = threadIdx.x;
  s[t] = partials[b * 256 + t];
  __syncthreads();
#pragma unroll
  for (int off = 128; off >= 1; off >>= 1) {
    if (t < off) s[t] += s[t + off];
    __syncthreads();
  }
  if (t == 0) probs[b] = s[0];
}

extern "C" void kernel_launch(void* const* d_in, const int* in_sizes, int n_in,
                              void* d_out, int out_size, void* d_ws, size_t ws_size,
                              hipStream_t stream) {
  const float* h      = (const float*)d_in[0];  // (32, 512, 320)
  const float* a_link = (const float*)d_in[1];  // (256, 64)
  const float* b_link = (const float*)d_in[2];  // (256, 64)
  // d_in[3] = rollout (unused: rollout path has no noise)

  float* ws = (float*)d_ws;
  float* Aout = (float*)d_out;                              // (32, 512, 512)
  float* probs = Aout + (size_t)BATCH * NNODE * NNODE;      // (32,)
  float* partials = ws + 2 * HI_ELEMS;                      // 8192 floats

  glcn_proj_kernel<<<256, 256, 0, stream>>>(h, a_link, b_link, ws);
  glcn_logits_kernel<<<1024, 256, 0, stream>>>(ws, Aout, partials);
  glcn_probs_kernel<<<32, 256, 0, stream>>>(partials, probs);
}